// Adagae_Processor_82042465288308
// MI455X (gfx1250) — compile-verified
//
#include <hip/hip_runtime.h>
#include <hip/hip_bf16.h>
#include <cmath>

typedef float v2f __attribute__((ext_vector_type(2)));
typedef float v8f __attribute__((ext_vector_type(8)));

#define HD    128   // hidden dim (x is [n,128])
#define TILE  64    // output tile per workgroup (64x64)
#define LDSTR 132   // padded LDS row stride in floats (bank-conflict-free v2f frags)

// ---------------- Pass 0: sumsq, h = x@W.T + b, labels, n_unknown ----------------
__global__ void precompute_kernel(const float* __restrict__ x,
                                  const float* __restrict__ ohm,
                                  const float* __restrict__ W,
                                  const float* __restrict__ bias,
                                  float* __restrict__ sumsq,
                                  float* __restrict__ h,
                                  int* __restrict__ labels,
                                  int* __restrict__ nunk,
                                  int n, int nclasses)
{
  const int wave = threadIdx.x >> 5;
  const int lane = threadIdx.x & 31;
  const int row  = blockIdx.x * (blockDim.x >> 5) + wave;
  if (row >= n) return;
  float s = 0.f, h0 = 0.f, h1 = 0.f;
#pragma unroll
  for (int c = 0; c < HD / 32; ++c) {
    float xv = x[(size_t)row * HD + lane + c * 32];
    s  += xv * xv;
    h0 += xv * W[lane + c * 32];
    h1 += xv * W[HD + lane + c * 32];
  }
#pragma unroll
  for (int off = 16; off > 0; off >>= 1) {   // wave32 reduction
    s  += __shfl_xor(s,  off);
    h0 += __shfl_xor(h0, off);
    h1 += __shfl_xor(h1, off);
  }
  if (lane == 0) {
    sumsq[row]     = s;
    h[row * 2 + 0] = h0 + bias[0];
    h[row * 2 + 1] = h1 + bias[1];
    int li = -1;
    for (int c = 0; c < nclasses; ++c)
      if (ohm[(size_t)row * nclasses + c] > 0.5f) li = c;
    labels[row] = li;
    if (li < 0) atomicAdd(nunk, 1);
  }
}

// ---------------- Pass 1: ED tile GEMM via V_WMMA_F32_16X16X4_F32 ----------------
// ed[i,j] = sqrt(max(s_i + s_j - 2*dot(x_i,x_j), 0)) written into adj region.
__global__ void __launch_bounds__(128)
ed_gemm_kernel(const float* __restrict__ x, const float* __restrict__ sumsq,
               float* __restrict__ ed, int n)
{
  __shared__ float As[TILE][LDSTR];
  __shared__ float Bs[TILE][LDSTR];
  const int t       = threadIdx.x;
  const int rowBase = blockIdx.y * TILE;
  const int colBase = blockIdx.x * TILE;

  { // stage 64x128 row-tile and col-tile of x into LDS (float4 coalesced)
    const int c4 = t & 31;   // float4 column (k = c4*4)
    const int r0 = t >> 5;   // 0..3
#pragma unroll
    for (int rr = 0; rr < 16; ++rr) {
      const int r = r0 + rr * 4;
      *(float4*)&As[r][c4 * 4] = *(const float4*)&x[(size_t)(rowBase + r) * HD + c4 * 4];
      *(float4*)&Bs[r][c4 * 4] = *(const float4*)&x[(size_t)(colBase + r) * HD + c4 * 4];
    }
  }
  __syncthreads();

  const int wave  = t >> 5;
  const int lane  = t & 31;
  const int mrow  = lane & 15;          // M (A) / N (B) index within tile
  const int khalf = (lane >> 4) * 2;    // lane half selects K pair
  const int mAdd  = (lane >> 4) * 8;    // C/D layout: lanes 16-31 hold M+8

  v8f acc[4] = {};                      // wave's 16-row strip x 4 col tiles

#if __has_builtin(__builtin_amdgcn_wmma_f32_16x16x4_f32)
  for (int k0 = 0; k0 < HD; k0 += 4) {
    v2f a  = *(const v2f*)&As[wave * 16 + mrow][k0 + khalf];
    v2f b0 = *(const v2f*)&Bs[ 0 + mrow][k0 + khalf];
    v2f b1 = *(const v2f*)&Bs[16 + mrow][k0 + khalf];
    v2f b2 = *(const v2f*)&Bs[32 + mrow][k0 + khalf];
    v2f b3 = *(const v2f*)&Bs[48 + mrow][k0 + khalf];
    acc[0] = __builtin_amdgcn_wmma_f32_16x16x4_f32(false, a, false, b0, (short)0, acc[0], false, false);
    acc[1] = __builtin_amdgcn_wmma_f32_16x16x4_f32(false, a, false, b1, (short)0, acc[1], false, false);
    acc[2] = __builtin_amdgcn_wmma_f32_16x16x4_f32(false, a, false, b2, (short)0, acc[2], false, false);
    acc[3] = __builtin_amdgcn_wmma_f32_16x16x4_f32(false, a, false, b3, (short)0, acc[3], false, false);
  }
#else
  // scalar fallback producing identical C/D register layout
#pragma unroll
  for (int tt = 0; tt < 4; ++tt)
#pragma unroll
    for (int r = 0; r < 8; ++r) {
      float s = 0.f;
      for (int k = 0; k < HD; ++k)
        s += As[wave * 16 + r + mAdd][k] * Bs[tt * 16 + mrow][k];
      acc[tt][r] = s;
    }
#endif

  // epilogue: distances
  const int gRow0 = rowBase + wave * 16;
  float si[8];
#pragma unroll
  for (int r = 0; r < 8; ++r) si[r] = sumsq[gRow0 + r + mAdd];
#pragma unroll
  for (int tt = 0; tt < 4; ++tt) {
    const int gj  = colBase + tt * 16 + mrow;
    const float sj = sumsq[gj];
#pragma unroll
    for (int r = 0; r < 8; ++r) {
      const int gi = gRow0 + r + mAdd;
      float d2 = si[r] + sj - 2.0f * acc[tt][r];
      ed[(size_t)gi * n + gj] = sqrtf(fmaxf(d2, 0.f));
    }
  }
}

// ---------------- Pass 2: exact (k+1)-smallest per row -> a, 1/denom ----------------
// One 256-thread block per row; n assumed 8192 (32 values / thread).
__global__ void __launch_bounds__(256)
select_kernel(const float* __restrict__ ed, const int* __restrict__ sparsPtr,
              float* __restrict__ avals, float* __restrict__ rden, int n)
{
  __shared__ unsigned long long red[256];
  const int row = blockIdx.x;
  const int tid = threadIdx.x;
  const float* rp = ed + (size_t)row * n;
  float v[32];
#pragma unroll
  for (int j = 0; j < 32; ++j) v[j] = rp[tid + j * 256];

  const int k = *sparsPtr;
  unsigned usedMask = 0u;
  float bsum = 0.f, aval = 0.f;

  for (int e = 0; e <= k; ++e) {
    unsigned best = 0xFFFFFFFFu, bidx = 0u;
#pragma unroll
    for (int j = 0; j < 32; ++j) {
      unsigned fb = __float_as_uint(v[j]);      // d >= 0: bit order == value order
      if ((((usedMask >> j) & 1u) == 0u) && fb < best) { best = fb; bidx = (unsigned)j; }
    }
    unsigned long long key =
        ((unsigned long long)best << 32) | (unsigned)(tid * 32 + (int)bidx);
    __syncthreads();
    red[tid] = key;
    __syncthreads();
    for (int s = 128; s > 0; s >>= 1) {
      if (tid < s) { unsigned long long o = red[tid + s]; if (o < red[tid]) red[tid] = o; }
      __syncthreads();
    }
    unsigned long long wk = red[0];
    const unsigned owner = (unsigned)(wk & 0xFFFFFFFFu);
    const float val = __uint_as_float((unsigned)(wk >> 32));
    if ((owner >> 5) == (unsigned)tid) usedMask |= 1u << (owner & 31u);
    if (e < k) bsum += val; else aval = val;
  }
  if (tid == 0) {
    float a = aval + 1e-10f;
    float denom = a * (float)k - bsum + 1e-10f;
    avals[row] = a;
    rden[row]  = 1.0f / denom;
  }
}

// ---------------- Pass 3: finalize adj in place ----------------
__global__ void adj_kernel(float* __restrict__ adj,
                           const float* __restrict__ avals,
                           const float* __restrict__ rden,
                           const int* __restrict__ labels,
                           const int* __restrict__ nunk, int n)
{
  const int row  = blockIdx.y;
  const int col0 = (blockIdx.x * blockDim.x + threadIdx.x) * 4;
  if (col0 >= n) return;
  const int   li = labels[row];
  const float ai = avals[row];
  const float ri = rden[row];
  const float fc = 1.0f / fmaxf((float)(*nunk), 1.0f);

  float4 d4 = *(float4*)&adj[(size_t)row * n + col0];
  float dv[4] = {d4.x, d4.y, d4.z, d4.w};
  float ov[4];
#pragma unroll
  for (int c = 0; c < 4; ++c) {
    const int col = col0 + c;
    const int lj  = labels[col];
    const float d = dv[c];
    float out;
    if (li >= 0 && lj >= 0) {
      out = (li == lj) ? 1.0f : 0.0f;                       // semantic (one-hot dot)
    } else if (li < 0 && lj < 0) {
      const float aj = avals[col];
      const float rj = rden[col];
      out = 0.5f * (fmaxf(ai - d, 0.f) * ri + fmaxf(aj - d, 0.f) * rj); // (A+A.T)/2
    } else {
      out = fc;
    }
    ov[c] = out;
  }
  float4 o4 = {ov[0], ov[1], ov[2], ov[3]};
  *(float4*)&adj[(size_t)row * n + col0] = o4;
}

// ---------------- Pass 4: g = tanh(adj @ h), one block per row ----------------
__global__ void __launch_bounds__(256)
g_kernel(const float* __restrict__ adj, const float* __restrict__ h,
         float* __restrict__ g, int n)
{
  __shared__ float r0[256], r1[256];
  const int row = blockIdx.x;
  const int tid = threadIdx.x;
  const float4* ar = (const float4*)(adj + (size_t)row * n);
  float a0 = 0.f, a1 = 0.f;
  const int n4 = n >> 2;
  for (int i = tid; i < n4; i += 256) {
    float4 w = ar[i];
    int c = i * 4;
    a0 += w.x * h[(c + 0) * 2]     + w.y * h[(c + 1) * 2]
        + w.z * h[(c + 2) * 2]     + w.w * h[(c + 3) * 2];
    a1 += w.x * h[(c + 0) * 2 + 1] + w.y * h[(c + 1) * 2 + 1]
        + w.z * h[(c + 2) * 2 + 1] + w.w * h[(c + 3) * 2 + 1];
  }
  r0[tid] = a0; r1[tid] = a1;
  __syncthreads();
  for (int s = 128; s > 0; s >>= 1) {
    if (tid < s) { r0[tid] += r0[tid + s]; r1[tid] += r1[tid + s]; }
    __syncthreads();
  }
  if (tid == 0) {
    g[row * 2 + 0] = tanhf(r0[0]);
    g[row * 2 + 1] = tanhf(r1[0]);
  }
}

// ---------------- Pass 5: centroids (single block, LDS atomics) ----------------
__global__ void centroid_kernel(const float* __restrict__ g, const int* __restrict__ labels,
                                float* __restrict__ cent, int n, int nclasses)
{
  __shared__ float cnt[32], agg0[32], agg1[32];
  const int tid = threadIdx.x;
  if (tid < 32) { cnt[tid] = 0.f; agg0[tid] = 0.f; agg1[tid] = 0.f; }
  __syncthreads();
  for (int r = tid; r < n; r += blockDim.x) {
    int li = labels[r];
    if (li >= 0) {
      atomicAdd(&cnt[li], 1.0f);
      atomicAdd(&agg0[li], g[r * 2 + 0]);
      atomicAdd(&agg1[li], g[r * 2 + 1]);
    }
  }
  __syncthreads();
  if (tid < nclasses) {
    float c = cnt[tid];
    cent[tid * 2 + 0] = (c > 0.f) ? agg0[tid] / c : 0.f;
    cent[tid * 2 + 1] = (c > 0.f) ? agg1[tid] / c : 0.f;
  }
}

// ---------------- Pass 6: scores = -cdist(g, centroids) ----------------
__global__ void scores_kernel(const float* __restrict__ g, const float* __restrict__ cent,
                              float* __restrict__ scores, int n, int nclasses)
{
  const int i = blockIdx.x * blockDim.x + threadIdx.x;
  if (i >= n) return;
  const float g0 = g[i * 2], g1 = g[i * 2 + 1];
  for (int c = 0; c < nclasses; ++c) {
    float dx = g0 - cent[c * 2], dy = g1 - cent[c * 2 + 1];
    scores[(size_t)i * nclasses + c] = -sqrtf(fmaxf(dx * dx + dy * dy, 0.f));
  }
}

extern "C" void kernel_launch(void* const* d_in, const int* in_sizes, int n_in,
                              void* d_out, int out_size, void* d_ws, size_t ws_size,
                              hipStream_t stream)
{
  const float* x     = (const float*)d_in[0];
  const float* ohm   = (const float*)d_in[1];
  const float* W     = (const float*)d_in[2];
  const float* bvec  = (const float*)d_in[3];
  const int*   spars = (const int*)d_in[4];

  const int H = in_sizes[2] / 2;          // 128
  const int n = in_sizes[0] / H;          // 8192
  const int C = in_sizes[1] / n;          // 10

  float* out    = (float*)d_out;
  float* scores = out;                     // [n, C]
  float* g      = out + (size_t)n * C;     // [n, 2]
  float* adj    = g + (size_t)n * 2;       // [n, n]  (used as ED scratch first)

  float* ws     = (float*)d_ws;
  float* sumsq  = ws;                                  // n
  float* h      = ws + (size_t)n;                      // 2n
  float* avals  = ws + (size_t)3 * n;                  // n
  float* rden   = ws + (size_t)4 * n;                  // n
  float* cent   = ws + (size_t)5 * n;                  // 2C (<64)
  int*   labels = (int*)(ws + (size_t)5 * n + 64);     // n ints
  int*   nunk   = (int*)(ws + (size_t)6 * n + 64);     // 1 int

  hipMemsetAsync(nunk, 0, sizeof(int), stream);

  precompute_kernel<<<n / 8, 256, 0, stream>>>(x, ohm, W, bvec, sumsq, h, labels, nunk, n, C);

  dim3 gGemm(n / TILE, n / TILE);
  ed_gemm_kernel<<<gGemm, 128, 0, stream>>>(x, sumsq, adj, n);

  select_kernel<<<n, 256, 0, stream>>>(adj, spars, avals, rden, n);

  dim3 gAdj(n / 1024, n);
  adj_kernel<<<gAdj, 256, 0, stream>>>(adj, avals, rden, labels, nunk, n);

  g_kernel<<<n, 256, 0, stream>>>(adj, h, g, n);

  centroid_kernel<<<1, 256, 0, stream>>>(g, labels, cent, n, C);

  scores_kernel<<<(n + 255) / 256, 256, 0, stream>>>(g, cent, scores, n, C);
}